// HyperSAGE_15255723835410
// MI455X (gfx1250) — compile-verified
//
#include <hip/hip_runtime.h>
#include <hip/hip_bf16.h>

typedef __attribute__((ext_vector_type(2))) float v2f;
typedef __attribute__((ext_vector_type(8))) float v8f;

#define MIN_V 1e-7f
#define MAX_V 10.0f

// ---------------------------------------------------------------------------
// Kernel 1: Hc = clip(H), P = Hc^2, Sig (accumulator) initialized to Hc.
// Vectorized float4: count4 = total_elems / 4.
// ---------------------------------------------------------------------------
__global__ void clip_pow_init(const float* __restrict__ X,
                              float* __restrict__ P,
                              float* __restrict__ Sig,
                              int count4) {
    int i = blockIdx.x * blockDim.x + threadIdx.x;
    if (i >= count4) return;
    float4 h = reinterpret_cast<const float4*>(X)[i];
    float4 hc, p;
    hc.x = fminf(fmaxf(h.x, MIN_V), MAX_V);
    hc.y = fminf(fmaxf(h.y, MIN_V), MAX_V);
    hc.z = fminf(fmaxf(h.z, MIN_V), MAX_V);
    hc.w = fminf(fmaxf(h.w, MIN_V), MAX_V);
    p.x = hc.x * hc.x; p.y = hc.y * hc.y; p.z = hc.z * hc.z; p.w = hc.w * hc.w;
    reinterpret_cast<float4*>(P)[i]   = p;
    reinterpret_cast<float4*>(Sig)[i] = hc;
}

// ---------------------------------------------------------------------------
// Kernel 2: per-(edge, feature) gather of the 16 member nodes' P values,
// power-mean-excluding-self, sqrt, scatter-add onto Sig with f32 atomics.
// D = feature dim; 256 threads/block -> EPB = 256/D edges per block.
// Gathers are coalesced across the feature dim (lane f, fixed node).
// P (102.4 MB for layer 1) is L2-resident on MI455X (192 MB L2).
// ---------------------------------------------------------------------------
template<int D>
__global__ __launch_bounds__(256) void edge_scatter(const float* __restrict__ P,
                                                    const int* __restrict__ edges,
                                                    float* __restrict__ Sig,
                                                    int E) {
    constexpr int S = 16;
    constexpr int EPB = 256 / D;            // edges per block
    __shared__ int sIdx[EPB * S];

    const int t  = threadIdx.x;
    const int e0 = blockIdx.x * EPB;

    if (t < EPB * S) {
        int e = e0 + t / S;
        sIdx[t] = (e < E) ? edges[e * S + (t % S)] : 0;
    }
    __syncthreads();

    const int le = t / D;                   // local edge
    const int f  = t % D;                   // feature
    if (e0 + le >= E) return;

    const int* idx = &sIdx[le * S];
    float p[S];
    float sum = 0.0f;
#pragma unroll
    for (int s = 0; s < S; ++s) {
        p[s] = P[idx[s] * D + f];
        sum += p[s];
    }
    const float inv = 1.0f / (float)(S - 1);
#pragma unroll
    for (int s = 0; s < S; ++s) {
        float c = sqrtf((sum - p[s]) * inv);       // contrib = neigh_mean^(1/2)
        unsafeAtomicAdd(&Sig[idx[s] * D + f], c);  // global_atomic_add_f32
    }
}

// ---------------------------------------------------------------------------
// Kernel 3: fused row-normalize + GEMM (+bias, optional ReLU) with
// V_WMMA_F32_16X16X4_F32. 8 waves/block, 16 rows per wave, K-loop step 4.
//   A fragment (16x4 f32): lane 0-15 = rows, koff = (lane>>4)*2
//   B fragment (4x16 f32): lane&15 = col, same koff split
//   C/D (16x16 f32, 8 VGPRs): row = v + (lane>>4)*8, col = lane&15
// Row normalization commutes with A@W -> applied in the epilogue
// (D[v] * rinv[row] + bias), keeping the K-loop pure ds_load + v_wmma.
// ---------------------------------------------------------------------------
template<int K, int NOUT, bool RELU>
__global__ __launch_bounds__(256) void gemm_norm(const float* __restrict__ Sig,
                                                 const float* __restrict__ W,
                                                 const float* __restrict__ b,
                                                 float* __restrict__ Out,
                                                 int Nrows) {
    constexpr int NT   = (NOUT + 15) / 16;  // N tiles (padded)
    constexpr int NT16 = NT * 16;
    constexpr int KP   = K + 4;             // LDS row pitch (bank-conflict pad)

    __shared__ float Wsh[K * NT16];
    __shared__ float bsh[NT16];
    __shared__ float Ash[8 * 16 * KP];
    __shared__ float rs[8 * 16];

    const int t = threadIdx.x;

    // stage W (zero-padded columns) and bias into LDS
    for (int i = t; i < K * NT16; i += 256) {
        int k = i / NT16, n = i % NT16;
        Wsh[i] = (n < NOUT) ? W[k * NOUT + n] : 0.0f;
    }
    if (t < NT16) bsh[t] = (t < NOUT) ? b[t] : 0.0f;
    __syncthreads();

    const int wave = t >> 5;
    const int lane = t & 31;
    const int r0   = blockIdx.x * 128 + wave * 16;   // Nrows % 16 == 0
    const bool valid = (r0 < Nrows);
    float* A = &Ash[wave * 16 * KP];

    // stage this wave's 16xK row tile (coalesced float4 global loads)
    if (valid) {
        const float* src = Sig + (size_t)r0 * K;
        for (int j = lane; j < 16 * K / 4; j += 32) {
            int row = (j * 4) / K;
            int col = (j * 4) % K;
            float4 v = reinterpret_cast<const float4*>(src)[j];
            *reinterpret_cast<float4*>(&A[row * KP + col]) = v;  // KP%4==0
        }
    }
    __syncthreads();

    // per-row sum -> reciprocal (rowsum >= d*1e-7 > 0, always finite)
    if (valid && lane < 16) {
        float s = 0.0f;
        for (int k = 0; k < K; ++k) s += A[lane * KP + k];
        rs[wave * 16 + lane] = 1.0f / s;
    }
    __syncthreads();
    if (!valid) return;

    const int m    = lane & 15;
    const int koff = (lane >> 4) * 2;
    const int rhi  = (lane >> 4) * 8;

    // per-output-row 1/rowsum, fetched once via conflict-free LDS broadcast
    float rowinv[8];
#pragma unroll
    for (int v = 0; v < 8; ++v) rowinv[v] = rs[wave * 16 + rhi + v];

    for (int nt = 0; nt < NT; ++nt) {
        v8f acc = {};
#pragma unroll
        for (int k = 0; k < K; k += 4) {
            v2f a, bf;
            a.x  = A[m * KP + k + koff];
            a.y  = A[m * KP + k + koff + 1];
            bf.x = Wsh[(k + koff)     * NT16 + nt * 16 + m];
            bf.y = Wsh[(k + koff + 1) * NT16 + nt * 16 + m];
            acc = __builtin_amdgcn_wmma_f32_16x16x4_f32(
                false, a, false, bf, (short)0, acc, false, false);
        }
        const int col = nt * 16 + m;
        if (col < NOUT) {
            const float bias = bsh[col];
#pragma unroll
            for (int v = 0; v < 8; ++v) {
                float val = fmaf(acc[v], rowinv[v], bias);  // diag(r)*(A@W)+b
                if (RELU) val = fmaxf(val, 0.0f);
                Out[(size_t)(r0 + rhi + v) * NOUT + col] = val;
            }
        }
    }
}

// ---------------------------------------------------------------------------
// Host-side orchestration (graph-capture safe: only async launches on stream)
// ---------------------------------------------------------------------------
extern "C" void kernel_launch(void* const* d_in, const int* in_sizes, int n_in,
                              void* d_out, int out_size, void* d_ws, size_t ws_size,
                              hipStream_t stream) {
    const float* H  = (const float*)d_in[0];   // [N,128]
    const float* W1 = (const float*)d_in[1];   // [128,16]
    const float* b1 = (const float*)d_in[2];   // [16]
    const float* W2 = (const float*)d_in[3];   // [16,40]
    const float* b2 = (const float*)d_in[4];   // [40]
    const int* edges = (const int*)d_in[5];    // [E,16]

    const int N = in_sizes[0] / 128;           // 200000
    const int E = in_sizes[5] / 16;            // 40000
    float* out = (float*)d_out;                // [N,40]

    // workspace layout (floats); P1 region is reused for X1/P2/S2 once dead.
    float* ws = (float*)d_ws;
    float* P1 = ws;                            // [N,128]  (dead after scatter1)
    float* S1 = ws + (size_t)N * 128;          // [N,128]  accumulator/new_signal
    float* X1 = ws;                            // [N,16]   layer-1 output (aliases P1)
    float* P2 = ws + (size_t)N * 16;           // [N,16]
    float* S2 = ws + (size_t)N * 32;           // [N,16]

    // ---- layer 1 (d=128 -> h=16, ReLU) ----
    {
        int c4 = N * 128 / 4;
        clip_pow_init<<<(c4 + 255) / 256, 256, 0, stream>>>(H, P1, S1, c4);
        edge_scatter<128><<<(E + 1) / 2, 256, 0, stream>>>(P1, edges, S1, E);
        gemm_norm<128, 16, true><<<(N + 127) / 128, 256, 0, stream>>>(S1, W1, b1, X1, N);
    }
    // ---- layer 2 (d=16 -> c=40) ----
    {
        int c4 = N * 16 / 4;
        clip_pow_init<<<(c4 + 255) / 256, 256, 0, stream>>>(X1, P2, S2, c4);
        edge_scatter<16><<<(E + 15) / 16, 256, 0, stream>>>(P2, edges, S2, E);
        gemm_norm<16, 40, false><<<(N + 127) / 128, 256, 0, stream>>>(S2, W2, b2, out, N);
    }
}